// EmbedLinear_80968723464887
// MI455X (gfx1250) — compile-verified
//
#include <hip/hip_runtime.h>
#include <cstdint>

// Problem constants from the reference: B=16384, F=4096 (copy width), C=4096 (gather width)
constexpr int F    = 4096;
constexpr int C    = 4096;
constexpr int OUTW = F + C;      // 8192 floats per output row
constexpr int R    = 4;          // rows staged per block (R*F*4 = 64 KB LDS)
constexpr int T    = 256;        // threads per block (8 waves of 32)
constexpr int NK   = (R * F) / (T * 4);   // 16 async b128 sweeps per lane
constexpr int NC   = C / (T * 4);         // 4 gather chunks per thread

// Native clang vector types (HIP's float4 is a struct; builtins need real vectors)
typedef float f32x4 __attribute__((ext_vector_type(4)));
typedef int   i32x4 __attribute__((ext_vector_type(4)));

// Template-recursive unroll: one global_load_async_to_lds_b128 per step, sharing
// the same address VGPRs; the signed-24-bit INST_OFFSET advances BOTH the global
// and LDS addresses by (K-1)*4096 bytes (CDNA5 ISA pseudocode).
template <int K> struct AsyncSweep {
    __device__ static void run(unsigned la, const float* ga) {
        AsyncSweep<K - 1>::run(la, ga);
        asm volatile("global_load_async_to_lds_b128 %0, %1, off offset:%2"
                     :
                     : "v"(la), "v"(ga), "n"((K - 1) * T * 16)
                     : "memory");
    }
};
template <> struct AsyncSweep<0> {
    __device__ static void run(unsigned, const float*) {}
};

__global__ __launch_bounds__(T)
void EmbedLinear_80968723464887_kernel(const float* __restrict__ in,
                                       const float* __restrict__ values,
                                       const int*   __restrict__ parents,
                                       float*       __restrict__ out,
                                       int B)
{
    __shared__ alignas(16) float rows[R * F];   // 64 KB: 4 full input rows

    const int  tid  = threadIdx.x;
    const long row0 = (long)blockIdx.x * R;
    if (row0 >= B) return;

    // ---- Phase A: async DMA of R input rows into LDS (16 B/lane/op, 16 ops) ----
    {
        const float* ga = in + row0 * (long)F + tid * 4;             // lane base, 16B apart
        unsigned     la = (unsigned)(uintptr_t)(&rows[0]) + (unsigned)tid * 16u;
        AsyncSweep<NK>::run(la, ga);                                  // 16x b128, imm offsets
    }

    // ---- Overlap: pull parents/values (int4/float4) while the DMA is in flight ----
    i32x4 p[NC];
    f32x4 w[NC];
    #pragma unroll
    for (int k = 0; k < NC; ++k) {
        const int c = k * T * 4 + tid * 4;
        p[k] = *(const i32x4*)(parents + c);
        w[k] = *(const f32x4*)(values  + c);
    }

    asm volatile("s_wait_asynccnt 0" ::: "memory");  // this wave's async ops done
    __syncthreads();                                  // all waves' rows visible in LDS

    // ---- Phase B: copy part  out[b, 0:F] = input[b, :]  (LDS -> NT b128 stores) ----
    #pragma unroll
    for (int r = 0; r < R; ++r) {
        const f32x4* lr   = (const f32x4*)(rows + r * F);
        f32x4*       drow = (f32x4*)(out + (row0 + r) * (long)OUTW);
        #pragma unroll
        for (int j = 0; j < F / 4 / T; ++j) {        // 4 iterations, static offsets
            const int idx = j * T + tid;
            __builtin_nontemporal_store(lr[idx], drow + idx);
        }
    }

    // ---- Phase C: gather part  out[b, F+c] = rows[b][parents[c]] * values[c] ----
    #pragma unroll
    for (int k = 0; k < NC; ++k) {
        const int c = k * T * 4 + tid * 4;
        #pragma unroll
        for (int r = 0; r < R; ++r) {
            const float* lr = rows + r * F;
            f32x4 o;
            o.x = lr[p[k].x] * w[k].x;
            o.y = lr[p[k].y] * w[k].y;
            o.z = lr[p[k].z] * w[k].z;
            o.w = lr[p[k].w] * w[k].w;
            f32x4* dst = (f32x4*)(out + (row0 + r) * (long)OUTW + F + c);
            __builtin_nontemporal_store(o, dst);
        }
    }
}

extern "C" void kernel_launch(void* const* d_in, const int* in_sizes, int n_in,
                              void* d_out, int out_size, void* d_ws, size_t ws_size,
                              hipStream_t stream)
{
    const float* in      = (const float*)d_in[0];   // [B, F] fp32
    const float* values  = (const float*)d_in[1];   // [C]    fp32
    const int*   parents = (const int*)  d_in[2];   // [C]    int32
    float*       out     = (float*)d_out;           // [B, F+C] fp32

    const int B = in_sizes[0] / F;                  // 16384
    dim3 grid((B + R - 1) / R);                     // 4096 blocks
    dim3 block(T);
    EmbedLinear_80968723464887_kernel<<<grid, block, 0, stream>>>(in, values, parents, out, B);
}